// SHEquivariantEncoder_10084583211580
// MI455X (gfx1250) — compile-verified
//
#include <hip/hip_runtime.h>
#include <hip/hip_bf16.h>

// ---------------- problem constants ----------------
#define LMAX   4
#define C      32
#define BATCH  64
#define NT     55
#define D_TOT  404480            // C*C*395
#define BD     (BATCH * D_TOT)   // 25,886,720
#define EPS    1e-6f

// per-l3 tables (l3 = 0..8)
__device__ __constant__ int BASE_L3[9]  = {0, 5120, 29696, 80896, 152576, 235520, 303104, 356352, 387072};
__device__ __constant__ int LEN_L3[9]   = {5120, 24576, 51200, 71680, 82944, 67584, 53248, 30720, 17408};
__device__ __constant__ int ROWBASE[9]  = {0, 5120, 13312, 23552, 33792, 43008, 49152, 53248, 55296};
#define NROWS 56320
// y (channel-mixed inputs) offsets per l, layout (b, o, m)
__device__ __constant__ int YOFF[5]     = {0, 2048, 8192, 18432, 32768};
#define YTOT 51200

// ---------------- workspace layout (floats) ----------------
#define OFF_Z     0
#define OFF_Y     25886720
#define OFF_S1    25937920
#define OFF_S2    25938496
#define OFF_ALPHA 25939072
#define OFF_BETA  25939648
#define OFF_MB    25940224
#define OFF_RSTD  25945344
#define OFF_LAT   26001664

typedef float v2f __attribute__((ext_vector_type(2)));
typedef float v8f __attribute__((ext_vector_type(8)));

// ---------------- kernel: zero the 64x128 latent accumulator ----------------
__global__ void k_init(float* __restrict__ lat) {
    int g = blockIdx.x * 256 + threadIdx.x;
    if (g < BATCH * 128) lat[g] = 0.0f;
}

// ---------------- kernel: y[l][b,o,m] = sum_i Wlin[l][o,i] x[l][b,i,m] (+bias for l=0) ----
__global__ void k_linmix(const float* __restrict__ x0, const float* __restrict__ x1,
                         const float* __restrict__ x2, const float* __restrict__ x3,
                         const float* __restrict__ x4, const float* __restrict__ Wlin,
                         const float* __restrict__ bias0, float* __restrict__ yall) {
    int idx = blockIdx.x * 256 + threadIdx.x;
    if (idx >= YTOT) return;
    int l = 0;
    #pragma unroll
    for (int q = 1; q < 5; q++) l += (idx >= YOFF[q]);
    int rem = idx - YOFF[l];
    int M = 2 * l + 1;
    int m  = rem % M;
    int t2 = rem / M;
    int o  = t2 & 31;
    int b  = t2 >> 5;
    const float* x = (l == 0) ? x0 : (l == 1) ? x1 : (l == 2) ? x2 : (l == 3) ? x3 : x4;
    const float* Wl = Wlin + l * (C * C) + o * C;
    float s = (l == 0) ? bias0[o] : 0.0f;
    #pragma unroll 8
    for (int i = 0; i < C; i++)
        s = fmaf(Wl[i], x[(size_t)b * C * M + i * M + m], s);
    yall[YOFF[l] + rem] = s;
}

// ---------------- tensor product inner body (K compile-time -> t[] in VGPRs) ----------------
template <int K>
__device__ __forceinline__ void tp_inner(const float* __restrict__ y1s, const float* __restrict__ y2s,
                                         const float* __restrict__ w3s, int m1, int m2,
                                         float* __restrict__ zrow, int i, int j) {
    float t[K];
    #pragma unroll
    for (int k = 0; k < K; k++) t[k] = 0.0f;
    for (int m = 0; m < m1; m++) {
        float a = y1s[i * m1 + m];
        for (int n = 0; n < m2; n++) {
            float c = a * y2s[j * m2 + n];
            const float* wp = &w3s[(m * m2 + n) * K];
            #pragma unroll
            for (int k = 0; k < K; k++) t[k] = fmaf(c, wp[k], t[k]);
        }
    }
    float ss = EPS;
    #pragma unroll
    for (int k = 0; k < K; k++) ss = fmaf(t[k], t[k], ss);
    float inv = rsqrtf(ss);
    #pragma unroll
    for (int k = 0; k < K; k++) zrow[k] = t[k] * inv;
}

// ---------------- kernel: tensor product + first row-normalization ----------------
// grid = BATCH*NT blocks, 256 threads. Each block: one (b, triple).
__global__ void k_tp(const float* __restrict__ yall, const float* __restrict__ w3j,
                     float* __restrict__ z) {
    int blk = blockIdx.x;
    int tt = blk % NT, b = blk / NT;
    // decode triple (uniform scalar loop)
    int l1 = 0, l2 = 0, l3 = 0, slot = 0;
    {
        int cnt[9] = {0, 0, 0, 0, 0, 0, 0, 0, 0};
        int idx = 0;
        for (int a = 0; a <= LMAX; a++)
            for (int b2 = a; b2 <= LMAX; b2++)
                for (int c = b2 - a; c <= a + b2; c++) {
                    if (idx == tt) { l1 = a; l2 = b2; l3 = c; slot = cnt[c]; }
                    cnt[c]++; idx++;
                }
    }
    int m1 = 2 * l1 + 1, m2 = 2 * l2 + 1, K = 2 * l3 + 1;
    __shared__ float y1s[288], y2s[288], w3s[1377];
    int tid = threadIdx.x;
    for (int q = tid; q < C * m1; q += 256) y1s[q] = yall[YOFF[l1] + b * C * m1 + q];
    for (int q = tid; q < C * m2; q += 256) y2s[q] = yall[YOFF[l2] + b * C * m2 + q];
    for (int q = tid; q < m1 * m2 * K; q += 256) {
        int k = q % K; int t2 = q / K; int n = t2 % m2; int m = t2 / m2;
        w3s[q] = w3j[(size_t)tt * 1377 + m * 153 + n * 17 + k];
    }
    __syncthreads();
    for (int p = tid; p < C * C; p += 256) {
        int i = p >> 5, j = p & 31;
        float* zrow = &z[(size_t)b * D_TOT + BASE_L3[l3] + (size_t)(slot * 1024 + p) * K];
        switch (l3) {
            case 0: tp_inner<1 >(y1s, y2s, w3s, m1, m2, zrow, i, j); break;
            case 1: tp_inner<3 >(y1s, y2s, w3s, m1, m2, zrow, i, j); break;
            case 2: tp_inner<5 >(y1s, y2s, w3s, m1, m2, zrow, i, j); break;
            case 3: tp_inner<7 >(y1s, y2s, w3s, m1, m2, zrow, i, j); break;
            case 4: tp_inner<9 >(y1s, y2s, w3s, m1, m2, zrow, i, j); break;
            case 5: tp_inner<11>(y1s, y2s, w3s, m1, m2, zrow, i, j); break;
            case 6: tp_inner<13>(y1s, y2s, w3s, m1, m2, zrow, i, j); break;
            case 7: tp_inner<15>(y1s, y2s, w3s, m1, m2, zrow, i, j); break;
            case 8: tp_inner<17>(y1s, y2s, w3s, m1, m2, zrow, i, j); break;
        }
    }
}

// ---------------- kernel: per-(b,l3) sum / sumsq over z ----------------
__global__ void k_red1(const float* __restrict__ z, float* __restrict__ S1, float* __restrict__ S2) {
    int blk = blockIdx.x;             // b*9 + l3
    int b = blk / 9, l3 = blk % 9;
    size_t base = (size_t)b * D_TOT + BASE_L3[l3];
    int len = LEN_L3[l3];
    int tid = threadIdx.x;
    float s = 0.0f, ss = 0.0f;
    for (int q = tid; q < len; q += 256) {
        float v = z[base + q];
        s += v; ss = fmaf(v, v, ss);
    }
    __shared__ float sh1[256], sh2[256];
    sh1[tid] = s; sh2[tid] = ss;
    __syncthreads();
    for (int off = 128; off > 0; off >>= 1) {
        if (tid < off) { sh1[tid] += sh1[tid + off]; sh2[tid] += sh2[tid + off]; }
        __syncthreads();
    }
    if (tid == 0) { S1[blk] = sh1[0]; S2[blk] = sh2[0]; }
}

// ---------------- kernel: fold global power scale + LN into one affine per (b,l3) ------------
__global__ void k_coef(const float* __restrict__ S1, const float* __restrict__ S2,
                       float* __restrict__ alpha, float* __restrict__ beta) {
    __shared__ float sh2[576];
    int tid = threadIdx.x;            // 576 threads = 64*9
    sh2[tid] = S2[tid];
    __syncthreads();
    int b = tid / 9, l3 = tid % 9;
    float power = 0.0f;
    #pragma unroll
    for (int q = 0; q < 9; q++) power += sh2[b * 9 + q];
    float scale = rsqrtf(power / (float)D_TOT + EPS);
    float N  = (float)LEN_L3[l3];
    float s1 = S1[tid] * scale;
    float s2 = sh2[tid] * scale * scale;
    float mu  = (l3 == 0) ? (s1 / N) : 0.0f;
    float var = s2 / N - mu * mu;
    float inv = rsqrtf(var + EPS);
    alpha[tid] = scale * inv;
    beta[tid]  = -mu * inv;
}

// ---------------- kernel: apply LN affine elementwise ----------------
__global__ void k_ln(float* __restrict__ z, const float* __restrict__ alpha,
                     const float* __restrict__ beta) {
    size_t g = (size_t)blockIdx.x * 256 + threadIdx.x;
    if (g >= (size_t)BD) return;
    int b = (int)(g / D_TOT);
    int pos = (int)(g % D_TOT);
    int l3 = 0;
    #pragma unroll
    for (int q = 1; q < 9; q++) l3 += (pos >= BASE_L3[q]);
    z[g] = fmaf(alpha[b * 9 + l3], z[g], beta[b * 9 + l3]);
}

// ---------------- kernel: BN batch-mean for l3==0 features (5120) ----------------
__global__ void k_bnmean(const float* __restrict__ z, float* __restrict__ mB) {
    int pos = blockIdx.x * 256 + threadIdx.x;
    if (pos >= 5120) return;
    float s = 0.0f;
    #pragma unroll 8
    for (int b = 0; b < BATCH; b++) s += z[(size_t)b * D_TOT + pos];
    mB[pos] = s * (1.0f / BATCH);
}

// ---------------- kernel: BN variance per row (axis 0 and 2) ----------------
__global__ void k_bnvar(const float* __restrict__ z, const float* __restrict__ mB,
                        float* __restrict__ rstd) {
    int r = blockIdx.x;               // row in [0, NROWS)
    int t = threadIdx.x;              // 64 threads = batch
    int l3 = 0;
    #pragma unroll
    for (int q = 1; q < 9; q++) l3 += (r >= ROWBASE[q]);
    int lr = r - ROWBASE[l3];
    int K = 2 * l3 + 1;
    size_t base = (size_t)t * D_TOT + BASE_L3[l3] + (size_t)lr * K;
    float mb = (l3 == 0) ? mB[lr] : 0.0f;
    float s = 0.0f;
    for (int k = 0; k < K; k++) {
        float v = z[base + k] - mb;
        s = fmaf(v, v, s);
    }
    __shared__ float sh[64];
    sh[t] = s;
    __syncthreads();
    for (int off = 32; off > 0; off >>= 1) {
        if (t < off) sh[t] += sh[t + off];
        __syncthreads();
    }
    if (t == 0) rstd[r] = rsqrtf(sh[0] / (64.0f * K) + EPS);
}

// ---------------- kernel: apply BN elementwise ----------------
__global__ void k_bnapply(float* __restrict__ z, const float* __restrict__ mB,
                          const float* __restrict__ rstd) {
    size_t g = (size_t)blockIdx.x * 256 + threadIdx.x;
    if (g >= (size_t)BD) return;
    int pos = (int)(g % D_TOT);
    int l3 = 0;
    #pragma unroll
    for (int q = 1; q < 9; q++) l3 += (pos >= BASE_L3[q]);
    int rem = pos - BASE_L3[l3];
    int K = 2 * l3 + 1;
    int lr = rem / K;
    float mb = (l3 == 0) ? mB[lr] : 0.0f;
    z[g] = (z[g] - mb) * rstd[ROWBASE[l3] + lr];
}

// ---------------- kernel: split-K FC, fp32 WMMA 16x16x4 + async-LDS double buffering --------
// lat[64][128] += feats[64, chunk] @ W[128, chunk]^T  (395 blocks x 1024-wide D chunks)
// Panels of K=64 are DMA'd straight into LDS with GLOBAL_LOAD_ASYNC_TO_LDS_B128
// (tracked by ASYNCcnt), double-buffered so HBM streaming overlaps the WMMAs.
#define LDA 68   // LDS row stride: 68 mod 64 = 4 -> conflict-free column reads
__global__ void k_fc(const float* __restrict__ feats, const float* __restrict__ Wg,
                     float* __restrict__ lat) {
    __shared__ float As[2][64 * LDA];    // [buf][b][d] panel, 64x64
    __shared__ float Ws[2][128 * LDA];   // [buf][o][d] panel, 128x64
    int tid = threadIdx.x;
    int lane = tid & 31, wv = tid >> 5;       // wave32: 8 waves
    int d0 = blockIdx.x * 1024;

    // Issue one panel's async loads: 3072 float4 (A:1024 + W:2048) -> 12 per thread/wave.
    auto issue_panel = [&](int pc, int buf) {
        int dp = d0 + pc * 64;
        for (int q = tid; q < 3072; q += 256) {
            const float* src;
            unsigned ldsoff;
            if (q < 1024) {                       // A panel: feats[bb][dp + c4*4 ..]
                int bb = q >> 4, c4 = q & 15;
                src = feats + (size_t)bb * D_TOT + dp + c4 * 4;
                ldsoff = (unsigned)(uintptr_t)&As[buf][bb * LDA + c4 * 4];
            } else {                              // W panel: W[oo][dp + c4*4 ..]
                int q2 = q - 1024;
                int oo = q2 >> 4, c4 = q2 & 15;
                src = Wg + (size_t)oo * D_TOT + dp + c4 * 4;
                ldsoff = (unsigned)(uintptr_t)&Ws[buf][oo * LDA + c4 * 4];
            }
            unsigned long long ga = (unsigned long long)(uintptr_t)src;
            // per-lane DMA: copies 16B from the global address in vaddr to the LDS address in vdst, tracked by ASYNCcnt
            asm volatile("global_load_async_to_lds_b128 %0, %1, off"
                         :: "v"(ldsoff), "v"(ga) : "memory");
        }
    };

    v8f acc[4];
    v8f zero8 = {0.f, 0.f, 0.f, 0.f, 0.f, 0.f, 0.f, 0.f};
    #pragma unroll
    for (int bt = 0; bt < 4; bt++) acc[bt] = zero8;
    const int n   = lane & 15;                // N col / A row within tile
    const int klo = (lane >> 4) * 2;          // K pair selector per ISA 16x4 fp32 layout
    const int otile = wv;                     // each wave owns one 16-wide O tile

    issue_panel(0, 0);                        // prime the pipeline
    for (int pc = 0; pc < 16; pc++) {
        int buf = pc & 1;
        if (pc < 15) {
            issue_panel(pc + 1, buf ^ 1);     // DMA next panel into the other buffer
            // 12 older async loads (panel pc) retire in order -> wait count 12
            asm volatile("s_wait_asynccnt 12" ::: "memory");
        } else {
            asm volatile("s_wait_asynccnt 0" ::: "memory");
        }
        __syncthreads();                      // all waves' panel pc now in LDS
        const float* Ab = &As[buf][0];
        const float* Wb = &Ws[buf][0];
        #pragma unroll
        for (int ks = 0; ks < 16; ks++) {
            int dk = ks * 4;
            v2f bf;
            bf.x = Wb[(otile * 16 + n) * LDA + dk + klo];
            bf.y = Wb[(otile * 16 + n) * LDA + dk + klo + 1];
            #pragma unroll
            for (int bt = 0; bt < 4; bt++) {
                v2f af;
                af.x = Ab[(bt * 16 + n) * LDA + dk + klo];
                af.y = Ab[(bt * 16 + n) * LDA + dk + klo + 1];
                acc[bt] = __builtin_amdgcn_wmma_f32_16x16x4_f32(
                    false, af, false, bf, (short)0, acc[bt], false, false);
            }
        }
        __syncthreads();                      // done reading buf; safe for DMA reuse
    }
    // scatter-accumulate: C/D layout: VGPR r, lane L -> M = r + (L>=16)*8, N = L&15
    #pragma unroll
    for (int bt = 0; bt < 4; bt++) {
        #pragma unroll
        for (int r = 0; r < 8; r++) {
            int row = bt * 16 + r + ((lane >> 4) * 8);
            int col = otile * 16 + n;
            atomicAdd(&lat[row * 128 + col], acc[bt][r]);
        }
    }
}

// ---------------- kernel: add bias, split mu/logvar into d_out ----------------
__global__ void k_final(const float* __restrict__ lat, const float* __restrict__ b_fc,
                        float* __restrict__ out) {
    int idx = blockIdx.x * 256 + threadIdx.x;
    if (idx >= BATCH * 128) return;
    int b = idx / 128, o = idx % 128;
    float v = lat[idx] + b_fc[o];
    if (o < 64) out[b * 64 + o] = v;                  // mu
    else        out[4096 + b * 64 + (o - 64)] = v;    // logvar
}

// ---------------- launch ----------------
extern "C" void kernel_launch(void* const* d_in, const int* in_sizes, int n_in,
                              void* d_out, int out_size, void* d_ws, size_t ws_size,
                              hipStream_t stream) {
    const float* x0    = (const float*)d_in[0];
    const float* x1    = (const float*)d_in[1];
    const float* x2    = (const float*)d_in[2];
    const float* x3    = (const float*)d_in[3];
    const float* x4    = (const float*)d_in[4];
    const float* Wlin  = (const float*)d_in[5];
    const float* bias0 = (const float*)d_in[6];
    const float* w3j   = (const float*)d_in[7];
    const float* W_fc  = (const float*)d_in[8];
    const float* b_fc  = (const float*)d_in[9];
    float* out = (float*)d_out;

    float* ws    = (float*)d_ws;
    float* z     = ws + OFF_Z;
    float* yall  = ws + OFF_Y;
    float* S1    = ws + OFF_S1;
    float* S2    = ws + OFF_S2;
    float* alpha = ws + OFF_ALPHA;
    float* beta  = ws + OFF_BETA;
    float* mB    = ws + OFF_MB;
    float* rstd  = ws + OFF_RSTD;
    float* lat   = ws + OFF_LAT;

    k_init   <<<32, 256, 0, stream>>>(lat);
    k_linmix <<<(YTOT + 255) / 256, 256, 0, stream>>>(x0, x1, x2, x3, x4, Wlin, bias0, yall);
    k_tp     <<<BATCH * NT, 256, 0, stream>>>(yall, w3j, z);
    k_red1   <<<BATCH * 9, 256, 0, stream>>>(z, S1, S2);
    k_coef   <<<1, 576, 0, stream>>>(S1, S2, alpha, beta);
    k_ln     <<<(BD + 255) / 256, 256, 0, stream>>>(z, alpha, beta);
    k_bnmean <<<20, 256, 0, stream>>>(z, mB);
    k_bnvar  <<<NROWS, 64, 0, stream>>>(z, mB, rstd);
    k_bnapply<<<(BD + 255) / 256, 256, 0, stream>>>(z, mB, rstd);
    k_fc     <<<D_TOT / 1024, 256, 0, stream>>>(z, W_fc, lat);
    k_final  <<<32, 256, 0, stream>>>(lat, b_fc, out);
}